// AmbientOrthogonalGroupBrownianMotion_72456098283813
// MI455X (gfx1250) — compile-verified
//
#include <hip/hip_runtime.h>
#include <stdint.h>

// Brownian motion on O(3): per-sample 3x3 state, 20 Euler steps with
// tangent projection + polar retraction (Newton-Schulz instead of SVD).
// Noise stream (189 MB) staged into LDS via the CDNA5 Tensor Data Mover,
// double-buffered and overlapped with the per-lane fp32 compute.
// All one-touch global traffic (x/t/out) uses non-temporal cache hints.

#define STEPS       20
#define BLOCK       256
#define TILE_FLOATS (BLOCK * 9)   // 2304 floats = 9216 bytes per step per block
#define NS_ITERS    6

typedef __attribute__((ext_vector_type(4))) unsigned int u32x4;
typedef __attribute__((ext_vector_type(8))) int          i32x8;
typedef __attribute__((ext_vector_type(4))) int          i32x4;

#if defined(__has_builtin)
#if __has_builtin(__builtin_amdgcn_tensor_load_to_lds)
#define HAVE_TDM 1
#endif
#if __has_builtin(__builtin_amdgcn_s_wait_tensorcnt)
#define HAVE_TENSORCNT 1
#endif
#endif

#ifdef HAVE_TDM
// Issue one TDM copy of TILE_FLOATS contiguous f32 from global -> LDS.
// D# layout per CDNA5 ISA ch.8 (group0 128b, group1 256b; groups 2/3 unused).
__device__ __forceinline__ void tdm_issue(uint32_t lds_addr, uint64_t gaddr) {
    u32x4 g0;
    g0.x = 1u;                                   // count=1 (valid descriptor)
    g0.y = lds_addr;                             // lds_addr [63:32]
    g0.z = (uint32_t)gaddr;                      // global_addr low
    g0.w = (uint32_t)((gaddr >> 32) & 0x01FFFFFFu) | (2u << 30); // ga hi | type=2

    i32x8 g1;
    g1[0] = (int)(2u << 16);                     // wg_mask=0, data_size=2 (4 bytes)
    g1[1] = (int)(((uint32_t)TILE_FLOATS & 0xFFFFu) << 16);    // tensor_dim0 lo16
    g1[2] = (int)(((uint32_t)TILE_FLOATS >> 16) | (1u << 16)); // dim0 hi | tensor_dim1=1
    g1[3] = (int)(((uint32_t)TILE_FLOATS) << 16);              // tile_dim0 = 2304
    g1[4] = 1;                                   // tile_dim1=1, tile_dim2=0
    g1[5] = (int)TILE_FLOATS;                    // tensor_dim0_stride lo32
    g1[6] = 0;                                   // stride hi / dim1_stride lo
    g1[7] = 0;

    i32x4 z4 = {0, 0, 0, 0};
#if defined(__clang_major__) && (__clang_major__ >= 23)
    i32x8 z8 = {0, 0, 0, 0, 0, 0, 0, 0};
    __builtin_amdgcn_tensor_load_to_lds(g0, g1, z4, z4, z8, 0);
#else
    __builtin_amdgcn_tensor_load_to_lds(g0, g1, z4, z4, 0);
#endif
}

__device__ __forceinline__ void tdm_wait() {
#ifdef HAVE_TENSORCNT
    __builtin_amdgcn_s_wait_tensorcnt(0);
#else
    asm volatile("s_wait_tensorcnt 0" ::: "memory");
#endif
}
#endif // HAVE_TDM

__global__ __launch_bounds__(BLOCK)
void obm_kernel(const float* __restrict__ x_in,
                const float* __restrict__ t_in,
                const float* __restrict__ noise,
                float* __restrict__ out,
                int nB)
{
    __shared__ float lds_noise[2][TILE_FLOATS];

    const int      tid        = (int)threadIdx.x;
    const uint32_t blockStart = blockIdx.x * BLOCK;
    const uint32_t gid        = blockStart + tid;

    // ---- load state + horizon (one-touch -> non-temporal) ----------------
    float x[9];
#pragma unroll
    for (int i = 0; i < 9; ++i)
        x[i] = __builtin_nontemporal_load(&x_in[(size_t)gid * 9 + i]);
    const float t   = __builtin_nontemporal_load(&t_in[gid]);
    const float sdt = sqrtf(t * (1.0f / (float)STEPS));

    // ---- prefetch step-0 noise tile into buffer 0 ------------------------
#ifdef HAVE_TDM
    if (tid < 32) {  // wave 0 drives the DMA; TENSORcnt is per-wave
        tdm_issue((uint32_t)(uintptr_t)&lds_noise[0][0],
                  (uint64_t)(uintptr_t)(noise + (size_t)blockStart * 9));
        tdm_wait();
    }
    __syncthreads();
#else
    for (int i = tid; i < TILE_FLOATS; i += BLOCK)
        lds_noise[0][i] = noise[(size_t)blockStart * 9 + i];
    __syncthreads();
#endif

    for (int s = 0; s < STEPS; ++s) {
        const int cur = s & 1;

        // kick off DMA for the next step's tile into the other buffer
        if (s + 1 < STEPS) {
#ifdef HAVE_TDM
            if (tid < 32) {
                tdm_issue((uint32_t)(uintptr_t)&lds_noise[cur ^ 1][0],
                          (uint64_t)(uintptr_t)
                              (noise + ((size_t)(s + 1) * nB + blockStart) * 9));
            }
#else
            for (int i = tid; i < TILE_FLOATS; i += BLOCK)
                lds_noise[cur ^ 1][i] =
                    noise[((size_t)(s + 1) * nB + blockStart) * 9 + i];
#endif
        }

        // ---- tangent-space increment: v = 0.5*(inc - x inc^T x) ----------
        float e[9];
#pragma unroll
        for (int i = 0; i < 9; ++i)
            e[i] = sdt * lds_noise[cur][tid * 9 + i];   // conflict-free: 9i mod 64

        float a[9];  // a = x * inc^T
#pragma unroll
        for (int r = 0; r < 3; ++r)
#pragma unroll
            for (int c = 0; c < 3; ++c)
                a[3 * r + c] = x[3 * r + 0] * e[3 * c + 0]
                             + x[3 * r + 1] * e[3 * c + 1]
                             + x[3 * r + 2] * e[3 * c + 2];

        float y[9];  // y = x + 0.5*(inc - a*x)
#pragma unroll
        for (int r = 0; r < 3; ++r)
#pragma unroll
            for (int c = 0; c < 3; ++c) {
                float w = a[3 * r + 0] * x[0 + c]
                        + a[3 * r + 1] * x[3 + c]
                        + a[3 * r + 2] * x[6 + c];
                y[3 * r + c] = x[3 * r + c] + 0.5f * (e[3 * r + c] - w);
            }

        // ---- polar retraction: Newton-Schulz y <- y(1.5I - 0.5 y^T y) ----
        // Y^T Y = I + v^T v (v tangent), sigma in [1, sqrt(3)) => quadratic
        // convergence to the SVD polar factor U V^T.
#pragma unroll
        for (int it = 0; it < NS_ITERS; ++it) {
            float s00 = y[0]*y[0] + y[3]*y[3] + y[6]*y[6];
            float s01 = y[0]*y[1] + y[3]*y[4] + y[6]*y[7];
            float s02 = y[0]*y[2] + y[3]*y[5] + y[6]*y[8];
            float s11 = y[1]*y[1] + y[4]*y[4] + y[7]*y[7];
            float s12 = y[1]*y[2] + y[4]*y[5] + y[7]*y[8];
            float s22 = y[2]*y[2] + y[5]*y[5] + y[8]*y[8];
            float t00 = 1.5f - 0.5f*s00, t01 = -0.5f*s01, t02 = -0.5f*s02;
            float t11 = 1.5f - 0.5f*s11, t12 = -0.5f*s12;
            float t22 = 1.5f - 0.5f*s22;
            float ny[9];
#pragma unroll
            for (int r = 0; r < 3; ++r) {
                ny[3*r+0] = y[3*r+0]*t00 + y[3*r+1]*t01 + y[3*r+2]*t02;
                ny[3*r+1] = y[3*r+0]*t01 + y[3*r+1]*t11 + y[3*r+2]*t12;
                ny[3*r+2] = y[3*r+0]*t02 + y[3*r+1]*t12 + y[3*r+2]*t22;
            }
#pragma unroll
            for (int i = 0; i < 9; ++i) y[i] = ny[i];
        }

#pragma unroll
        for (int i = 0; i < 9; ++i) x[i] = y[i];

        // publish next buffer / retire current one before it is overwritten
#ifdef HAVE_TDM
        if (s + 1 < STEPS && tid < 32) tdm_wait();
#endif
        __syncthreads();
    }

    // ---- one-touch output: non-temporal (write-rinse through L2) ---------
#pragma unroll
    for (int i = 0; i < 9; ++i)
        __builtin_nontemporal_store(x[i], &out[(size_t)gid * 9 + i]);
}

extern "C" void kernel_launch(void* const* d_in, const int* in_sizes, int n_in,
                              void* d_out, int out_size, void* d_ws, size_t ws_size,
                              hipStream_t stream) {
    const float* x     = (const float*)d_in[0];   // [B,3,3]
    const float* t     = (const float*)d_in[1];   // [B,1]
    const float* noise = (const float*)d_in[2];   // [STEPS,B,3,3]
    float*       out   = (float*)d_out;           // [B,3,3]
    const int nB = in_sizes[1];                   // B = 262144 (t element count)
    const int blocks = nB / BLOCK;
    obm_kernel<<<blocks, BLOCK, 0, stream>>>(x, t, noise, out, nB);
}